// RandomPropagate_47897475285096
// MI455X (gfx1250) — compile-verified
//
#include <hip/hip_runtime.h>
#include <stdint.h>

// ---------------- problem constants (match reference) ----------------
#define H_DIM       2048
#define TILE        64
#define GRID_T      (H_DIM / TILE)     // 32
#define PITCH       (TILE + 2)         // 66 (tile + halo)
#define CELLS       (PITCH * PITCH)    // 4356
#define NUM_SPREADS 100
#define MIN_T       1e-4f
#define THR         0.05f

struct Ctrl { unsigned changed; int empty; int done; int pad; };

#ifndef __has_builtin
#define __has_builtin(x) 0
#endif
#if __has_builtin(__builtin_amdgcn_tensor_load_to_lds)
#define USE_TDM 1
#else
#define USE_TDM 0
#warning "CDNA5 TDM builtin __builtin_amdgcn_tensor_load_to_lds NOT available; using per-lane fallback tile load"
#endif
#if USE_TDM && __has_builtin(__builtin_amdgcn_tensor_store_from_lds)
#define USE_TDM_STORE 1
#else
#define USE_TDM_STORE 0
#if USE_TDM
#warning "CDNA5 TDM builtin __builtin_amdgcn_tensor_store_from_lds NOT available; using per-lane stores"
#endif
#endif

typedef unsigned int u32x4 __attribute__((ext_vector_type(4)));
typedef int          i32x4 __attribute__((ext_vector_type(4)));
typedef int          i32x8 __attribute__((ext_vector_type(8)));

// ---------------- counter-based RNG (stand-in for threefry) ----------------
__device__ __forceinline__ unsigned mix32(unsigned x) {
  x ^= x >> 16; x *= 0x7feb352du;
  x ^= x >> 15; x *= 0x846ca68bu;
  x ^= x >> 16;
  return x;
}
__device__ __forceinline__ float uniform01(unsigned salt, unsigned idx) {
  unsigned h = mix32(idx ^ (0x9e3779b9u * (salt + 1u)));
  return (float)(h >> 8) * (1.0f / 16777216.0f);
}

#if USE_TDM
struct TdmDesc { u32x4 g0; i32x8 g1; i32x4 g2; i32x4 g3; };

// Common D# group0 + tensor dims (2048 x 2048 f32, row stride 2048 elems).
__device__ __forceinline__ void tdm_fill_common(TdmDesc& d, const void* gptr,
                                                unsigned lds_byte, unsigned flags1) {
  unsigned long long ga = (unsigned long long)gptr;
  d.g0.x = 1u;                                           // count=1 (valid)
  d.g0.y = lds_byte;                                     // LDS byte address
  d.g0.z = (unsigned)(ga & 0xffffffffu);                 // global addr lo
  d.g0.w = (unsigned)((ga >> 32) & 0x01ffffffu) | (2u << 30);  // addr hi | type=2
  d.g1.s0 = (int)((2u << 16) | flags1);                  // data_size=4B | flags
  d.g1.s1 = (int)(((unsigned)H_DIM & 0xffffu) << 16);    // tensor_dim0 lo16
  d.g1.s2 = (int)(((unsigned)H_DIM & 0xffffu) << 16);    // dim0 hi16(0) | dim1 lo16
  d.g1.s5 = H_DIM;                                       // tensor_dim0_stride lo32
  d.g1.s6 = 0;
  d.g1.s7 = 0;
  d.g3 = (i32x4){0, 0, 0, 0};
}

// Plain 2-D tile: w x h elements, LDS pitch == w (contiguous rows).
__device__ __forceinline__ TdmDesc tdm_make_2d(const void* gptr, unsigned lds_byte,
                                               int w, int h) {
  TdmDesc d;
  tdm_fill_common(d, gptr, lds_byte, 0u);
  d.g1.s3 = (int)((unsigned)w << 16);                    // dim1 hi16(0) | tile_dim0=w
  d.g1.s4 = h;                                           // tile_dim1=h, tile_dim2=0
  d.g2 = (i32x4){0, 0, 0, 0};
  return d;
}

// Iterated rows: w elems/row, h rows, LDS pitch = lds_pitch elems (for halo tile).
__device__ __forceinline__ TdmDesc tdm_make_rows(const void* gptr, unsigned lds_byte,
                                                 int w, int h, int lds_pitch) {
  TdmDesc d;
  tdm_fill_common(d, gptr, lds_byte, 1u << 19);          // iterate_enable
  d.g1.s3 = (int)((unsigned)w << 16);                    // tile_dim0 = w
  d.g1.s4 = 1;                                           // tile_dim1 = 1 row/iter
  d.g2.x = 0;
  d.g2.y = lds_pitch;                                    // lds_addr_increment
  d.g2.z = H_DIM;                                        // global_addr_increment (1 row)
  d.g2.w = (int)(((unsigned)(h - 1)) << 16);             // iterate_count = h-1
  return d;
}

#if __clang_major__ >= 23
#define TDM_LOAD(d)  do { i32x8 z8_ = {0,0,0,0,0,0,0,0}; \
  __builtin_amdgcn_tensor_load_to_lds((d).g0,(d).g1,(d).g2,(d).g3,z8_,0); } while (0)
#define TDM_STORE(d) do { i32x8 z8_ = {0,0,0,0,0,0,0,0}; \
  __builtin_amdgcn_tensor_store_from_lds((d).g0,(d).g1,(d).g2,(d).g3,z8_,0); } while (0)
#else
#define TDM_LOAD(d)  __builtin_amdgcn_tensor_load_to_lds((d).g0,(d).g1,(d).g2,(d).g3,0)
#define TDM_STORE(d) __builtin_amdgcn_tensor_store_from_lds((d).g0,(d).g1,(d).g2,(d).g3,0)
#endif
#endif  // USE_TDM

// ---------------- init: x0 = seed * habitat * goodness; reset ctrl ----------
__global__ __launch_bounds__(256)
void rp_init_kernel(const float* __restrict__ seed,
                    const float* __restrict__ habitat,
                    const float* __restrict__ goodness,
                    float* __restrict__ x0, Ctrl* __restrict__ ctrl) {
  const int i = (int)(blockIdx.x * blockDim.x + threadIdx.x);
  if (i < H_DIM * H_DIM) x0[i] = seed[i] * habitat[i] * goodness[i];
  if (i == 0) { ctrl->changed = 0u; ctrl->empty = 0; ctrl->done = 0; }
}

// ---------------- one spread step ------------------------------------------
__global__ __launch_bounds__(256)
void rp_step_kernel(const float* __restrict__ src, float* __restrict__ dst,
                    const float* __restrict__ goodness,
                    Ctrl* __restrict__ ctrl, int step) {
  __shared__ float xt[CELLS];            // x tile + halo (TDM destination)
  __shared__ float yt[CELLS];            // jitter, then jittered tile, then output
  __shared__ float gt[TILE * TILE];      // goodness tile (TDM destination)
  __shared__ unsigned sChanged;

  const int tid = (int)threadIdx.x;
  const int bx = (int)blockIdx.x, by = (int)blockIdx.y;
  const int lx = tid & (TILE - 1);
  const int ty = tid >> 6;               // 0..3 -> 16 output rows each

  // Early-exit emulation: once done, carry state through unchanged.
  if (ctrl->done) {
    const int gx = bx * TILE + lx;
    for (int k = 0; k < 16; ++k) {
      const int gy = by * TILE + ty * 16 + k;
      const int gidx = gy * H_DIM + gx;
      dst[gidx] = src[gidx];
    }
    return;
  }

  if (tid == 0) sChanged = 0u;

  const int gx0 = bx * TILE - 1;
  const int gy0 = by * TILE - 1;
  const unsigned salt1 = 2u * (unsigned)step;

#if USE_TDM
  // Border blocks: pre-zero xt so the out-of-image halo reads as 0
  // (equivalent to -inf padding since all values are >= 0).
  const bool edge = (bx == 0) | (by == 0) | (bx == GRID_T - 1) | (by == GRID_T - 1);
  if (edge) {
    for (int c = tid; c < CELLS; c += 256) xt[c] = 0.0f;
  }
  __syncthreads();                       // zero-fill complete before DMA writes LDS

  // Issue both tile DMAs (one wave) -- do NOT wait yet.
  if (tid == 0) {
    const int cx0 = gx0 < 0 ? 0 : gx0;
    const int cy0 = gy0 < 0 ? 0 : gy0;
    int cx1 = gx0 + PITCH; if (cx1 > H_DIM) cx1 = H_DIM;
    int cy1 = gy0 + PITCH; if (cy1 > H_DIM) cy1 = H_DIM;
    const int w = cx1 - cx0, h = cy1 - cy0;
    const int sx = cx0 - gx0, sy = cy0 - gy0;
    const unsigned xt_base = (unsigned)(size_t)(void*)xt;  // low 32 bits = LDS offset
    const unsigned gt_base = (unsigned)(size_t)(void*)gt;
    TdmDesc dx = tdm_make_rows(src + (size_t)cy0 * H_DIM + cx0,
                               xt_base + (unsigned)(sy * PITCH + sx) * 4u,
                               w, h, PITCH);
    TDM_LOAD(dx);
    TdmDesc dg = tdm_make_2d(goodness + (size_t)(by * TILE) * H_DIM + bx * TILE,
                             gt_base, TILE, TILE);
    TDM_LOAD(dg);
  }

  // Overlap the DMAs with the RNG: jitter factor depends only on (step, pixel).
  for (int c = tid; c < CELLS; c += 256) {
    const int cy = c / PITCH, cx2 = c - cy * PITCH;
    const unsigned key = (unsigned)((gy0 + cy) * H_DIM + (gx0 + cx2));
    yt[c] = (1.0f - MIN_T) + MIN_T * uniform01(salt1, key);
  }

  if (tid == 0) __builtin_amdgcn_s_wait_tensorcnt(0);  // both DMAs landed
  __syncthreads();                                     // publish xt/gt to all waves

  // y = x * jitter
  for (int c = tid; c < CELLS; c += 256) yt[c] *= xt[c];
  __syncthreads();
#else
  for (int c = tid; c < CELLS; c += 256) {
    const int cy = c / PITCH, cx2 = c - cy * PITCH;
    const int gy = gy0 + cy, gx = gx0 + cx2;
    float v = 0.0f;
    if ((unsigned)gy < (unsigned)H_DIM && (unsigned)gx < (unsigned)H_DIM)
      v = src[gy * H_DIM + gx];
    xt[c] = v;
    const unsigned key = (unsigned)(gy * H_DIM + gx);
    yt[c] = v * ((1.0f - MIN_T) + MIN_T * uniform01(salt1, key));
  }
  __syncthreads();
#endif

  // 3x3 maxpool with a rolling 3-row window; then mask/threshold/update.
  const unsigned salt2 = salt1 + 1u;
  const int cx = lx + 1;
  const float* ycol = &yt[lx];           // window columns lx..lx+2
  const int r = ty * 16;
  float rm0 = fmaxf(fmaxf(ycol[r * PITCH], ycol[r * PITCH + 1]), ycol[r * PITCH + 2]);
  float rm1 = fmaxf(fmaxf(ycol[(r + 1) * PITCH], ycol[(r + 1) * PITCH + 1]),
                    ycol[(r + 1) * PITCH + 2]);
  unsigned localChanged = 0u;
  float xn[16];

  #pragma unroll
  for (int k = 0; k < 16; ++k) {
    const int rb = (r + 2 + k) * PITCH;
    const float rm2 = fmaxf(fmaxf(ycol[rb], ycol[rb + 1]), ycol[rb + 2]);
    const float m = fmaxf(fmaxf(rm0, rm1), rm2);

    const int ly = ty * 16 + k;
    const int gidx = (by * TILE + ly) * H_DIM + bx * TILE + lx;

#if USE_TDM
    const float g = gt[ly * TILE + lx];
#else
    const float g = goodness[gidx];
#endif
    const float u2 = uniform01(salt2, (unsigned)gidx);
    const float y4 = (u2 > 0.5f) ? (m * g) : 0.0f;   // coin toss * goodness
    const float xx = xt[(ly + 1) * PITCH + cx];
    const float delta = y4 - xx;
    xn[k] = (delta > THR) ? y4 : xx;                 // thresholded max-update
    localChanged += (delta != 0.0f) ? 1u : 0u;
#if !USE_TDM_STORE
    dst[gidx] = xn[k];
#endif
    rm0 = rm1; rm1 = rm2;
  }

  if (localChanged) atomicAdd(&sChanged, localChanged);

#if USE_TDM_STORE
  __syncthreads();                       // all neighbor reads of yt/xt finished
  // Pack results into LDS (reuse yt's first 64x64 slots, pitch = TILE).
  #pragma unroll
  for (int k = 0; k < 16; ++k) yt[(ty * 16 + k) * TILE + lx] = xn[k];
  __syncthreads();                       // packed tile visible to TDM engine
  if (tid == 0) {
    if (sChanged) atomicAdd(&ctrl->changed, sChanged);
    const unsigned yt_base = (unsigned)(size_t)(void*)yt;
    TdmDesc ds = tdm_make_2d(dst + (size_t)(by * TILE) * H_DIM + bx * TILE,
                             yt_base, TILE, TILE);
    TDM_STORE(ds);                       // drained by implicit wait at ENDPGM
  }
#else
  __syncthreads();
  if (tid == 0 && sChanged) atomicAdd(&ctrl->changed, sChanged);
#endif
}

// ---------------- per-step scalar bookkeeping ------------------------------
__global__ void rp_finalize_kernel(Ctrl* __restrict__ ctrl) {
  if (threadIdx.x == 0 && blockIdx.x == 0) {
    if (!ctrl->done) {
      if (ctrl->changed == 0u) ctrl->empty += 1; else ctrl->empty = 0;
      if (ctrl->empty > 4) ctrl->done = 1;
    }
    ctrl->changed = 0u;
  }
}

// ---------------- host-side launch -----------------------------------------
extern "C" void kernel_launch(void* const* d_in, const int* in_sizes, int n_in,
                              void* d_out, int out_size, void* d_ws, size_t ws_size,
                              hipStream_t stream) {
  (void)in_sizes; (void)n_in; (void)out_size; (void)ws_size;

  const float* seed     = (const float*)d_in[0];
  const float* habitat  = (const float*)d_in[1];
  const float* goodness = (const float*)d_in[2];

  float* buf0 = (float*)d_out;                                   // ping
  float* buf1 = (float*)d_ws;                                    // pong
  Ctrl*  ctrl = (Ctrl*)((char*)d_ws + (size_t)H_DIM * H_DIM * sizeof(float));

  const int n = H_DIM * H_DIM;
  rp_init_kernel<<<(n + 255) / 256, 256, 0, stream>>>(seed, habitat, goodness,
                                                      buf0, ctrl);

  dim3 grid(GRID_T, GRID_T);
  for (int i = 0; i < NUM_SPREADS; ++i) {
    const float* src = (i & 1) ? buf1 : buf0;
    float*       dst = (i & 1) ? buf0 : buf1;
    rp_step_kernel<<<grid, 256, 0, stream>>>(src, dst, goodness, ctrl, i);
    rp_finalize_kernel<<<1, 1, 0, stream>>>(ctrl);
  }
  // NUM_SPREADS is even -> final state lands back in buf0 == d_out.
}